// GraphNeuralNetwork_1864015807124
// MI455X (gfx1250) — compile-verified
//
#include <hip/hip_runtime.h>
#include <hip/hip_bf16.h>

typedef __attribute__((ext_vector_type(2))) float v2f;
typedef __attribute__((ext_vector_type(4))) float v4f;
typedef __attribute__((ext_vector_type(8))) float v8f;

#define DHID 64
#define BN_EPS 1e-5f

static __device__ __forceinline__ void atomicAddF(float* p, float v) {
    __hip_atomic_fetch_add(p, v, __ATOMIC_RELAXED, __HIP_MEMORY_SCOPE_AGENT);
}

// ---------------- degree / edge normalization ----------------

__global__ void k_init_deg(float* deg, int n) {
    int i = blockIdx.x * blockDim.x + threadIdx.x;
    if (i < n) deg[i] = 1.0f;   // self-loop fill_value = 1
}

__global__ void k_deg_accum(float* deg, const int* dst, const float* ew, int e) {
    int i = blockIdx.x * blockDim.x + threadIdx.x;
    if (i < e) atomicAddF(&deg[dst[i]], ew[i]);
}

__global__ void k_deg_finalize(float* dinv, int n) {
    int i = blockIdx.x * blockDim.x + threadIdx.x;
    if (i < n) dinv[i] = rsqrtf(dinv[i]);
}

__global__ void k_edge_norm(float* norm, const float* ew, const int* src,
                            const int* dst, const float* dinv, int e) {
    int i = blockIdx.x * blockDim.x + threadIdx.x;
    if (i < e) norm[i] = ew[i] * dinv[src[i]] * dinv[dst[i]];
}

// ---------------- fp32 WMMA GEMM + fused self-loop epilogue ----------------
// out[M,64] = A[M,K] @ W[K,64] + bias ; agg[M,64] = out * dinv^2 (self-loop term).
// One wave computes a 16x64 strip with V_WMMA_F32_16X16X4_F32.
// A fragment (16x4): lanes 0-15 rows M=0..15 hold {K=k0,k0+1}; lanes 16-31 hold {k0+2,k0+3}.
// B fragment (4x16): lane r = column; v[0]={K=k0 | k0+2}, v[1]={K=k0+1 | k0+3}.
// C/D: v[i] -> row 8*half + i, col n0 + r.
__global__ __launch_bounds__(256) void k_gemm_wmma(const float* __restrict__ A,
                                                   const float* __restrict__ W,
                                                   const float* __restrict__ bias,
                                                   const float* __restrict__ dinv,
                                                   float* __restrict__ out,
                                                   float* __restrict__ agg,
                                                   int M, int K) {
    const int wave  = blockIdx.x * (blockDim.x >> 5) + (threadIdx.x >> 5);
    const int lane  = threadIdx.x & 31;
    const int half  = lane >> 4;     // 0 or 1
    const int r     = lane & 15;
    const int mbase = wave * 16;
    if (mbase >= M) return;          // wave-uniform guard (EXEC stays all-ones)

    const float* arow = A + (size_t)(mbase + r) * K;

    v8f acc0 = {}, acc1 = {}, acc2 = {}, acc3 = {};
    for (int k0 = 0; k0 < K; k0 += 4) {
        v2f a;
        a.x = arow[k0 + 2 * half];
        a.y = arow[k0 + 2 * half + 1];

        const float* wr = W + (size_t)(k0 + 2 * half) * DHID + r;
        v2f b0; b0.x = wr[0];  b0.y = wr[DHID];
        acc0 = __builtin_amdgcn_wmma_f32_16x16x4_f32(false, a, false, b0, (short)0, acc0, false, false);
        v2f b1; b1.x = wr[16]; b1.y = wr[DHID + 16];
        acc1 = __builtin_amdgcn_wmma_f32_16x16x4_f32(false, a, false, b1, (short)0, acc1, false, false);
        v2f b2; b2.x = wr[32]; b2.y = wr[DHID + 32];
        acc2 = __builtin_amdgcn_wmma_f32_16x16x4_f32(false, a, false, b2, (short)0, acc2, false, false);
        v2f b3; b3.x = wr[48]; b3.y = wr[DHID + 48];
        acc3 = __builtin_amdgcn_wmma_f32_16x16x4_f32(false, a, false, b3, (short)0, acc3, false, false);
    }

    // self-loop coefficients for the 8 rows this lane stores
    float di2[8];
#pragma unroll
    for (int i = 0; i < 8; ++i) {
        float dv = dinv[mbase + half * 8 + i];
        di2[i] = dv * dv;
    }

    v8f accs[4] = {acc0, acc1, acc2, acc3};
#pragma unroll
    for (int nt = 0; nt < 4; ++nt) {
        const int col = nt * 16 + r;
        const float bv = bias[col];
#pragma unroll
        for (int i = 0; i < 8; ++i) {
            const size_t idx = (size_t)(mbase + half * 8 + i) * DHID + col;
            const float v = accs[nt][i] + bv;
            out[idx] = v;
            agg[idx] = v * di2[i];
        }
    }
}

// ---------------- scatter aggregation: agg[dst,:] += norm[e] * lin[src,:] ----------------
// float4 per thread: 16 threads cover one edge's 64 features (b128 gather + 4 f32 atomics).
__global__ void k_scatter(float* __restrict__ agg, const float* __restrict__ lin,
                          const float* __restrict__ norm,
                          const int* __restrict__ src, const int* __restrict__ dst,
                          int e) {
    long long i = (long long)blockIdx.x * blockDim.x + threadIdx.x;
    if (i < (long long)e * 16) {
        int ed = (int)(i >> 4);
        int q  = (int)(i & 15);
        const v4f* linv = (const v4f*)lin;
        v4f v = linv[(size_t)src[ed] * 16 + q] * norm[ed];
        float* base = agg + (size_t)dst[ed] * DHID + q * 4;
        atomicAddF(base + 0, v.x);
        atomicAddF(base + 1, v.y);
        atomicAddF(base + 2, v.z);
        atomicAddF(base + 3, v.w);
    }
}

// ---------------- batch norm ----------------

__global__ void k_zero(float* p, int n) {
    int i = blockIdx.x * blockDim.x + threadIdx.x;
    if (i < n) p[i] = 0.0f;
}

__global__ __launch_bounds__(256) void k_bn_stats(const float* __restrict__ agg,
                                                  float* __restrict__ stats, int n) {
    const int q  = threadIdx.x & 15;   // feature quad (4 floats)
    const int rl = threadIdx.x >> 4;   // 0..15 rows per block-iteration
    const v4f* aggv = (const v4f*)agg;
    v4f s = {}, s2 = {};
    for (int node = blockIdx.x * 16 + rl; node < n; node += gridDim.x * 16) {
        v4f v = aggv[(size_t)node * 16 + q];
        s += v; s2 += v * v;
    }
    __shared__ v4f sh[2][16][16];
    sh[0][rl][q] = s; sh[1][rl][q] = s2;
    __syncthreads();
    if (rl == 0) {
        v4f ts = {}, ts2 = {};
#pragma unroll
        for (int j = 0; j < 16; ++j) { ts += sh[0][j][q]; ts2 += sh[1][j][q]; }
        atomicAddF(&stats[q * 4 + 0], ts.x);
        atomicAddF(&stats[q * 4 + 1], ts.y);
        atomicAddF(&stats[q * 4 + 2], ts.z);
        atomicAddF(&stats[q * 4 + 3], ts.w);
        atomicAddF(&stats[DHID + q * 4 + 0], ts2.x);
        atomicAddF(&stats[DHID + q * 4 + 1], ts2.y);
        atomicAddF(&stats[DHID + q * 4 + 2], ts2.z);
        atomicAddF(&stats[DHID + q * 4 + 3], ts2.w);
    }
}

// scale = g * rsqrt(var+eps); shift = be - mean*scale
__global__ void k_bn_finalize(float* scsh, const float* stats,
                              const float* g, const float* be, int n) {
    int d = threadIdx.x;
    if (d < DHID) {
        float inv_n = 1.0f / (float)n;
        float mean = stats[d] * inv_n;
        float var  = stats[DHID + d] * inv_n - mean * mean;
        float sc   = g[d] * rsqrtf(var + BN_EPS);
        scsh[d]        = sc;
        scsh[DHID + d] = be[d] - mean * sc;
    }
}

static __device__ __forceinline__ v4f bn_relu4(v4f v, const float* scsh, int q) {
    const v4f sc = ((const v4f*)scsh)[q];
    const v4f sf = ((const v4f*)scsh)[16 + q];
    v4f r;
    r.x = fmaxf(v.x * sc.x + sf.x, 0.0f);
    r.y = fmaxf(v.y * sc.y + sf.y, 0.0f);
    r.z = fmaxf(v.z * sc.z + sf.z, 0.0f);
    r.w = fmaxf(v.w * sc.w + sf.w, 0.0f);
    return r;
}

// layers 0,1: h = relu(bn(agg))
__global__ void k_bn_apply_relu(float* __restrict__ h, const float* __restrict__ agg,
                                const float* __restrict__ scsh, int n) {
    int i = blockIdx.x * blockDim.x + threadIdx.x;
    if (i < n * 16) {
        int q = i & 15;
        ((v4f*)h)[i] = bn_relu4(((const v4f*)agg)[i], scsh, q);
    }
}

// layer 2 fused with global-mean-pool accumulation (skips materializing H)
__global__ void k_bn_relu_pool(float* __restrict__ psum, float* __restrict__ pcnt,
                               const float* __restrict__ agg, const float* __restrict__ scsh,
                               const int* __restrict__ batch, int n) {
    int i = blockIdx.x * blockDim.x + threadIdx.x;
    if (i < n * 16) {
        int node = i >> 4;
        int q    = i & 15;
        v4f r = bn_relu4(((const v4f*)agg)[i], scsh, q);
        int g = batch[node];
        float* base = psum + (size_t)g * DHID + q * 4;
        atomicAddF(base + 0, r.x);
        atomicAddF(base + 1, r.y);
        atomicAddF(base + 2, r.z);
        atomicAddF(base + 3, r.w);
        if (q == 0) atomicAddF(&pcnt[g], 1.0f);
    }
}

__global__ __launch_bounds__(64) void k_pool_out(float* out, const float* psum,
                                                 const float* pcnt, const float* ow,
                                                 const float* ob, int gcount) {
    int g = blockIdx.x;
    int d = threadIdx.x;
    __shared__ float red[DHID];
    float c = fmaxf(pcnt[g], 1.0f);
    red[d] = (psum[(size_t)g * DHID + d] / c) * ow[d];
    __syncthreads();
#pragma unroll
    for (int s = 32; s > 0; s >>= 1) {
        if (d < s) red[d] += red[d + s];
        __syncthreads();
    }
    if (d == 0) out[g] = red[0] + ob[0];
}

// ---------------- host launcher ----------------

extern "C" void kernel_launch(void* const* d_in, const int* in_sizes, int n_in,
                              void* d_out, int out_size, void* d_ws, size_t ws_size,
                              hipStream_t stream) {
    const float* x     = (const float*)d_in[0];
    const int*   ei    = (const int*)d_in[1];
    const float* ew    = (const float*)d_in[2];
    const int*   batch = (const int*)d_in[3];
    const float* w[3]  = {(const float*)d_in[4], (const float*)d_in[8],  (const float*)d_in[12]};
    const float* bb[3] = {(const float*)d_in[5], (const float*)d_in[9],  (const float*)d_in[13]};
    const float* gg[3] = {(const float*)d_in[6], (const float*)d_in[10], (const float*)d_in[14]};
    const float* be[3] = {(const float*)d_in[7], (const float*)d_in[11], (const float*)d_in[15]};
    const float* out_w = (const float*)d_in[16];
    const float* out_b = (const float*)d_in[17];
    float* out = (float*)d_out;

    const int N   = in_sizes[3];
    const int E   = in_sizes[2];
    const int DIN = in_sizes[0] / N;
    const int G   = out_size;

    const int* src = ei;
    const int* dst = ei + E;

    // workspace layout (floats)
    float* ws    = (float*)d_ws;
    float* dinv  = ws;
    float* norm  = dinv + N;
    float* L     = norm + E;                 // GEMM output [N,64]
    float* AGG   = L + (size_t)N * DHID;     // aggregate   [N,64]
    float* H     = AGG + (size_t)N * DHID;   // hidden      [N,64]
    float* stats = H + (size_t)N * DHID;     // [128] sum/sumsq
    float* scsh  = stats + 128;              // [128] scale/shift
    float* psum  = scsh + 128;               // [G,64]
    float* pcnt  = psum + (size_t)G * DHID;  // [G]

    const int T = 256;
    auto cdiv = [](long long a, long long b) { return (int)((a + b - 1) / b); };

    // edge normalization + pool-buffer init
    k_init_deg<<<cdiv(N, T), T, 0, stream>>>(dinv, N);
    k_deg_accum<<<cdiv(E, T), T, 0, stream>>>(dinv, dst, ew, E);
    k_deg_finalize<<<cdiv(N, T), T, 0, stream>>>(dinv, N);
    k_edge_norm<<<cdiv(E, T), T, 0, stream>>>(norm, ew, src, dst, dinv, E);
    k_zero<<<cdiv((long long)G * DHID + G, T), T, 0, stream>>>(psum, G * DHID + G);

    const int mtiles      = cdiv(N, 16);
    const int gemm_blocks = cdiv(mtiles, T / 32);
    const long long nVec4 = (long long)N * 16;   // float4 elements per [N,64] tensor
    const long long eVec4 = (long long)E * 16;

    const float* hin = x;
    int K = DIN;
    for (int layer = 0; layer < 3; ++layer) {
        k_gemm_wmma<<<gemm_blocks, T, 0, stream>>>(hin, w[layer], bb[layer], dinv, L, AGG, N, K);
        k_scatter<<<cdiv(eVec4, T), T, 0, stream>>>(AGG, L, norm, src, dst, E);
        k_zero<<<1, 128, 0, stream>>>(stats, 128);
        k_bn_stats<<<768, T, 0, stream>>>(AGG, stats, N);
        k_bn_finalize<<<1, DHID, 0, stream>>>(scsh, stats, gg[layer], be[layer], N);
        if (layer < 2) {
            k_bn_apply_relu<<<cdiv(nVec4, T), T, 0, stream>>>(H, AGG, scsh, N);
            hin = H;
            K = DHID;
        } else {
            k_bn_relu_pool<<<cdiv(nVec4, T), T, 0, stream>>>(psum, pcnt, AGG, scsh, batch, N);
        }
    }

    k_pool_out<<<G, DHID, 0, stream>>>(out, psum, pcnt, out_w, out_b, G);
}